// VectorQuantizer_3178275799663
// MI455X (gfx1250) — compile-verified
//
#include <hip/hip_runtime.h>
#include <math.h>

typedef float v8f __attribute__((ext_vector_type(8)));
typedef float v2f __attribute__((ext_vector_type(2)));

#define N_ROWS 65536
#define K_CODES 1024
#define D_DIM 256
#define LDS_STRIDE 260   // 256 + 4 pad: 16B-aligned rows, 4-bank row skew -> conflict-free b64 reads

// ---------------------------------------------------------------------------
// CDNA5 async global->LDS copy (ASYNCcnt-tracked), inline asm per bridge doc.
// GVS form: mem = SADDR(64) + VADDR(32) ; LDS dest byte address in first VGPR.
// ---------------------------------------------------------------------------
__device__ __forceinline__ void async_ld_b128(unsigned lds_addr, unsigned voff,
                                              unsigned long long sbase) {
  asm volatile("global_load_async_to_lds_b128 %0, %1, %2"
               :: "v"(lds_addr), "v"(voff), "s"(sbase) : "memory");
}
__device__ __forceinline__ void wait_async0() {
  asm volatile("s_wait_asynccnt 0" ::: "memory");
}

// ---------------------------------------------------------------------------
// zero the accumulators / histogram (graph-replay safe: done every launch)
// ---------------------------------------------------------------------------
__global__ void vq_init(float* __restrict__ counts, double* __restrict__ sums) {
  int t = blockIdx.x * blockDim.x + threadIdx.x;
  if (t < K_CODES) counts[t] = 0.0f;
  if (t < 2) sums[t] = 0.0;
}

// ---------------------------------------------------------------------------
// wsq[k] = |w_k|^2   (one block of 256 threads per codebook row)
// ---------------------------------------------------------------------------
__global__ void vq_wsq(const float* __restrict__ w, float* __restrict__ wsq) {
  __shared__ float red[8];
  const int row = blockIdx.x;
  float v = w[(size_t)row * D_DIM + threadIdx.x];
  float s = v * v;
  #pragma unroll
  for (int off = 16; off > 0; off >>= 1) s += __shfl_xor(s, off, 32);
  const int wid = threadIdx.x >> 5, lane = threadIdx.x & 31;
  if (lane == 0) red[wid] = s;
  __syncthreads();
  if (threadIdx.x == 0) {
    float tot = 0.0f;
    #pragma unroll
    for (int i = 0; i < 8; ++i) tot += red[i];
    wsq[row] = tot;
  }
}

// ---------------------------------------------------------------------------
// Main kernel: score[i][k] = wsq[k] - 2 * x_i . w_k  via V_WMMA_F32_16X16X4_F32,
// running argmin over k.  Block = 256 threads (8 waves), 128 rows of x in LDS,
// 16-code weight tiles DOUBLE-BUFFERED through LDS with async-to-LDS copies:
// tile ct+1 streams in (ASYNCcnt) while the 64 WMMAs of tile ct execute.
// ---------------------------------------------------------------------------
__global__ void __launch_bounds__(256)
vq_argmin(const float* __restrict__ x, const float* __restrict__ w,
          const float* __restrict__ wsq, int* __restrict__ idx,
          float* __restrict__ idx_f) {
  extern __shared__ float smem[];
  float* xs   = smem;                        // 128 x LDS_STRIDE
  float* wb   = xs + 128 * LDS_STRIDE;       // 2 x (16 x LDS_STRIDE) double buffer
  float* wsqs = wb + 2 * 16 * LDS_STRIDE;    // 1024

  const int tid  = threadIdx.x;
  const int lane = tid & 31;
  const int wave = tid >> 5;
  const int half = lane >> 4;                // 0: rows r, 1: rows r+8 (C layout)
  const int l15  = lane & 15;
  const int rowBase = blockIdx.x * 128;

  const unsigned long long xg = (unsigned long long)(uintptr_t)x;
  const unsigned long long wg = (unsigned long long)(uintptr_t)w;
  const unsigned xs_lds = (unsigned)(uintptr_t)xs;
  const unsigned wb_lds = (unsigned)(uintptr_t)wb;
  const unsigned WBUF_BYTES = 16u * LDS_STRIDE * 4u;

  // ---- stage x tile (128x256 f32) via async-to-LDS, b128 per lane ----
  #pragma unroll
  for (int i = 0; i < 32; ++i) {
    const int item = tid + 256 * i;          // 8192 float4 items
    const int r = item >> 6, c4 = item & 63;
    async_ld_b128(xs_lds + (unsigned)(r * LDS_STRIDE + c4 * 4) * 4u,
                  (unsigned)(((rowBase + r) * D_DIM + c4 * 4) * 4), xg);
  }
  // wsq -> LDS (tiny, plain stores)
  #pragma unroll
  for (int i = 0; i < 4; ++i) wsqs[tid + 256 * i] = wsq[tid + 256 * i];

  // per-thread offsets for the 16x256 weight tile (4 b128 items each)
  unsigned wloff[4], wgoff[4];
  #pragma unroll
  for (int i = 0; i < 4; ++i) {
    const int item = tid + 256 * i;          // 1024 float4 items
    const int r = item >> 6, c4 = item & 63;
    wloff[i] = (unsigned)(r * LDS_STRIDE + c4 * 4) * 4u;
    wgoff[i] = (unsigned)((r * D_DIM + c4 * 4) * 4);
  }

  // ---- prologue: issue tile 0 into buffer 0 ----
  #pragma unroll
  for (int i = 0; i < 4; ++i) async_ld_b128(wb_lds + wloff[i], wgoff[i], wg);
  wait_async0();
  __syncthreads();

  float bestS[8];
  int   bestI[8];
  #pragma unroll
  for (int r = 0; r < 8; ++r) { bestS[r] = __builtin_inff(); bestI[r] = 0; }

  const int m0 = wave * 16;
  const float* ap = &xs[(m0 + l15) * LDS_STRIDE + 2 * half];   // A: (M=l15, K=2*half..+1)

  for (int ct = 0; ct < K_CODES / 16; ++ct) {
    // issue next tile into the other buffer while we compute this one
    if (ct + 1 < K_CODES / 16) {
      const unsigned dst = wb_lds + (unsigned)((ct + 1) & 1) * WBUF_BYTES;
      const unsigned long long src = wg + (unsigned long long)(ct + 1) * 16u * D_DIM * 4u;
      #pragma unroll
      for (int i = 0; i < 4; ++i) async_ld_b128(dst + wloff[i], wgoff[i], src);
    }

    // acc starts at -wsq[n]/2 so that  -2*acc_final = wsq[n] - 2 * x.w
    const float cinit = -0.5f * wsqs[ct * 16 + l15];
    v8f acc;
    #pragma unroll
    for (int r = 0; r < 8; ++r) acc[r] = cinit;

    const float* bp = &wb[(ct & 1) * 16 * LDS_STRIDE + l15 * LDS_STRIDE + 2 * half];
    #pragma unroll 8
    for (int kk = 0; kk < D_DIM; kk += 4) {
      v2f a = *(const v2f*)(ap + kk);
      v2f b = *(const v2f*)(bp + kk);
      acc = __builtin_amdgcn_wmma_f32_16x16x4_f32(false, a, false, b,
                                                  (short)0, acc, false, false);
    }

    const int n = ct * 16 + l15;
    #pragma unroll
    for (int r = 0; r < 8; ++r) {
      const float s = -2.0f * acc[r];
      if (s < bestS[r]) { bestS[r] = s; bestI[r] = n; }
    }

    // my async issues (tile ct+1) landed + everyone done reading buffers
    wait_async0();
    __syncthreads();
  }

  // reduce across the 16 lanes of each half (rows 0-7 live in lanes 0-15,
  // rows 8-15 in lanes 16-31); tie-break on smaller index (argmin semantics)
  #pragma unroll
  for (int off = 1; off <= 8; off <<= 1) {
    #pragma unroll
    for (int r = 0; r < 8; ++r) {
      const float os = __shfl_xor(bestS[r], off, 32);
      const int   oi = __shfl_xor(bestI[r], off, 32);
      if (os < bestS[r] || (os == bestS[r] && oi < bestI[r])) {
        bestS[r] = os; bestI[r] = oi;
      }
    }
  }
  if (l15 == 0) {
    #pragma unroll
    for (int r = 0; r < 8; ++r) {
      const int grow = rowBase + m0 + r + 8 * half;
      idx[grow]   = bestI[r];
      idx_f[grow] = (float)bestI[r];
    }
  }
}

// ---------------------------------------------------------------------------
// quantized_out = weight[idx]  (== straight-through forward), accumulate
// sum((q-x)^2) and the code histogram.  One 256-thread block per row.
// ---------------------------------------------------------------------------
__global__ void vq_gather(const float* __restrict__ x, const float* __restrict__ w,
                          const int* __restrict__ idx, float* __restrict__ qout,
                          float* __restrict__ counts, double* __restrict__ sums) {
  __shared__ float red[8];
  const int row = blockIdx.x;
  const int k = idx[row];
  const float q  = w[(size_t)k * D_DIM + threadIdx.x];
  const float xv = x[(size_t)row * D_DIM + threadIdx.x];
  qout[(size_t)row * D_DIM + threadIdx.x] = q;
  const float d = q - xv;
  float s = d * d;
  #pragma unroll
  for (int off = 16; off > 0; off >>= 1) s += __shfl_xor(s, off, 32);
  const int wid = threadIdx.x >> 5, lane = threadIdx.x & 31;
  if (lane == 0) red[wid] = s;
  __syncthreads();
  if (threadIdx.x == 0) {
    float tot = 0.0f;
    #pragma unroll
    for (int i = 0; i < 8; ++i) tot += red[i];
    atomicAdd(&sums[0], (double)tot);
    atomicAdd(&counts[k], 1.0f);
  }
}

// ---------------------------------------------------------------------------
// util_loss = mean |counts - N/K|   (single block, 1024 threads)
// ---------------------------------------------------------------------------
__global__ void vq_util(const float* __restrict__ counts, float* __restrict__ out_util) {
  __shared__ float red[32];
  float s = fabsf(counts[threadIdx.x] - (float)N_ROWS / (float)K_CODES);
  #pragma unroll
  for (int off = 16; off > 0; off >>= 1) s += __shfl_xor(s, off, 32);
  const int wid = threadIdx.x >> 5, lane = threadIdx.x & 31;
  if (lane == 0) red[wid] = s;
  __syncthreads();
  if (threadIdx.x == 0) {
    float tot = 0.0f;
    #pragma unroll
    for (int i = 0; i < 32; ++i) tot += red[i];
    out_util[0] = tot / (float)K_CODES;
  }
}

// ---------------------------------------------------------------------------
// Pairwise codebook distances (upper triangle): one wave per 16x16 tile of
// the K x K matrix, same f32 WMMA trick, sqrt + masked sum.
// ---------------------------------------------------------------------------
__global__ void __launch_bounds__(256)
vq_compact(const float* __restrict__ w, const float* __restrict__ wsq,
           double* __restrict__ sums) {
  const int tid  = threadIdx.x;
  const int lane = tid & 31;
  const int wave = tid >> 5;
  const int tile = blockIdx.x * 8 + wave;     // 4096 tiles (64 x 64)
  const int tr = tile >> 6, tc = tile & 63;
  const int m0 = tr * 16, n0 = tc * 16;
  if (m0 >= n0 + 15) return;                  // wave-uniform: no i<j element in tile

  const int half = lane >> 4;
  const int l15  = lane & 15;

  const float cinit = -0.5f * wsq[n0 + l15];
  v8f acc;
  #pragma unroll
  for (int r = 0; r < 8; ++r) acc[r] = cinit;

  const float* ap = &w[(size_t)(m0 + l15) * D_DIM + 2 * half];
  const float* bp = &w[(size_t)(n0 + l15) * D_DIM + 2 * half];
  #pragma unroll 8
  for (int kk = 0; kk < D_DIM; kk += 4) {
    v2f a = *(const v2f*)(ap + kk);
    v2f b = *(const v2f*)(bp + kk);
    acc = __builtin_amdgcn_wmma_f32_16x16x4_f32(false, a, false, b,
                                                (short)0, acc, false, false);
  }

  float sum = 0.0f;
  #pragma unroll
  for (int r = 0; r < 8; ++r) {
    const int i = m0 + r + 8 * half;
    const int j = n0 + l15;
    // -2*acc = wsq[j] - 2 w_i.w_j  ->  pd2 = wsq[i] + wsq[j] - 2 w_i.w_j
    const float pd2 = wsq[i] - 2.0f * acc[r];
    if (i < j) sum += sqrtf(fmaxf(pd2, 1e-12f));
  }
  #pragma unroll
  for (int off = 16; off > 0; off >>= 1) sum += __shfl_xor(sum, off, 32);
  if (lane == 0) atomicAdd(&sums[1], (double)sum);
}

// ---------------------------------------------------------------------------
// final scalars
// ---------------------------------------------------------------------------
__global__ void vq_final(const double* __restrict__ sums, float* __restrict__ out) {
  const double n_elems = (double)N_ROWS * (double)D_DIM;
  const double n_pairs = (double)K_CODES * (double)(K_CODES - 1) * 0.5;
  out[(size_t)N_ROWS * D_DIM]     = (float)(1.25 * sums[0] / n_elems);  // quant_loss
  out[(size_t)N_ROWS * D_DIM + 2] = (float)(2.0 * sums[1] / n_pairs);   // compact_loss
}

// ---------------------------------------------------------------------------
extern "C" void kernel_launch(void* const* d_in, const int* in_sizes, int n_in,
                              void* d_out, int out_size, void* d_ws, size_t ws_size,
                              hipStream_t stream) {
  (void)in_sizes; (void)n_in; (void)out_size; (void)ws_size;
  const float* x = (const float*)d_in[0];   // [N, D]
  const float* w = (const float*)d_in[1];   // [K, D]
  float* out = (float*)d_out;

  char* ws = (char*)d_ws;
  int*    idx    = (int*)ws;                       // N ints
  float*  wsq    = (float*)(ws + 262144);          // K floats
  float*  counts = (float*)(ws + 266240);          // K floats
  double* sums   = (double*)(ws + 270336);         // [quantSum, compactSum]

  float* q_out = out;                               // [N, D]
  float* scal  = out + (size_t)N_ROWS * D_DIM;      // quant, util, compact
  float* idx_f = scal + 3;                          // [N] indices as float

  vq_init<<<4, 256, 0, stream>>>(counts, sums);
  vq_wsq<<<K_CODES, 256, 0, stream>>>(w, wsq);

  const size_t shmem = (size_t)(128 * LDS_STRIDE + 2 * 16 * LDS_STRIDE + 1024) * sizeof(float);
  vq_argmin<<<N_ROWS / 128, 256, shmem, stream>>>(x, w, wsq, idx, idx_f);

  vq_gather<<<N_ROWS, 256, 0, stream>>>(x, w, idx, q_out, counts, sums);
  vq_util<<<1, 1024, 0, stream>>>(counts, scal + 1);
  vq_compact<<<(64 * 64) / 8, 256, 0, stream>>>(w, wsq, sums);
  vq_final<<<1, 1, 0, stream>>>(sums, out);
}